// CRF_63565515981286
// MI455X (gfx1250) — compile-verified
//
#include <hip/hip_runtime.h>
#include <hip/hip_bf16.h>

#define TAGS 48
#define SEQ 512
#define NB 16            // batch columns per workgroup (WMMA N)
#define ES 52            // E LDS row stride (floats): even (float2 align), banks spread
#define QS 52            // Qt LDS row stride
#define SS 20            // score / feats LDS row stride
#define NEGV -10000.0f
#define START_IDX 45
#define STOP_IDX 46
#define LN2F 0.6931471805599453f

typedef float v2f __attribute__((ext_vector_type(2)));
typedef float v8f __attribute__((ext_vector_type(8)));

// raw v_log_f32 (log2) * ln2 — input clamped to a normal float, no denorm fixup needed
__device__ __forceinline__ float fast_log(float x) {
    return __builtin_amdgcn_logf(x) * LN2F;
}
// single v_med3_f32 clamp (input is always finite here)
__device__ __forceinline__ float clamp_lo(float x) {
    return __builtin_amdgcn_fmed3f(x, 1e-30f, 3.0e38f);
}

__global__ __launch_bounds__(96) void crf_fwd_wmma(
    const float* __restrict__ feats,   // [B, SEQ, TAGS]
    const float* __restrict__ trans,   // [TAGS, TAGS]
    const float* __restrict__ masks,   // [B, SEQ]
    float* __restrict__ out)           // [B]
{
    __shared__ float E[TAGS * ES];          // exp(trans)
    __shared__ float Qt[NB * QS];           // Q transposed: [n][j]
    __shared__ float Sc[TAGS * SS];         // score: [j][n]
    __shared__ float Fb[2][TAGS * SS];      // feats tile double buffer: [i][n]
    __shared__ float Pm[3 * NB];            // per-wave partial column maxes

    const int t  = threadIdx.x;
    const int w  = t >> 5;                  // wave id 0..2 (M block)
    const int l  = t & 31;                  // lane
    const int n  = t & 15;                  // batch column (== lane&15 for every wave)
    const int hi = (l >> 4);                // 0 / 1 (half-wave)
    const int nbase = blockIdx.x * NB;
    const int i0 = (t >> 4) * 8;            // this thread's 8-row slice: 0,8,...,40

    // ---- one-time init: E = exp(trans); initial score; partial maxes ----
    for (int idx = t; idx < TAGS * TAGS; idx += 96) {
        int i = idx / TAGS, j = idx - i * TAGS;
        E[i * ES + j] = __expf(trans[idx]);
    }
    for (int idx = t; idx < TAGS * NB; idx += 96) {
        int i = idx >> 4, c = idx & 15;
        Sc[i * SS + c] = (i == START_IDX) ? 0.0f : NEGV;
    }
    if (t < 3 * NB) Pm[t] = ((t >> 4) == (START_IDX / 16)) ? 0.0f : NEGV;

    // ---- prefetch feats(s=0) and mask(s=0) ----
    const float* fptr = feats + (size_t)(nbase + n) * SEQ * TAGS;
    const float* mptr = masks + (size_t)(nbase + n) * SEQ;
    float4 fr0, fr1; float mreg;
    {
        const float4* p = (const float4*)(fptr + i0);   // 16B aligned: row=192B, i0*4 mod 16 == 0
        fr0 = p[0]; fr1 = p[1];
        mreg = mptr[0];
    }

    for (int s = 0; s < SEQ; ++s) {
        const int buf = s & 1;
        __syncthreads();                                 // (a) Sc/Pm from prev epilogue ready

        // stage feats(s) regs -> LDS (read after barrier (b))
        {
            float fv[8] = {fr0.x, fr0.y, fr0.z, fr0.w, fr1.x, fr1.y, fr1.z, fr1.w};
            #pragma unroll
            for (int k = 0; k < 8; ++k) Fb[buf][(i0 + k) * SS + n] = fv[k];
        }
        const float mcur = mreg;

        // issue prefetch for step s+1 (overlaps with exp/WMMA below)
        if (s + 1 < SEQ) {
            const float4* p = (const float4*)(fptr + (size_t)(s + 1) * TAGS + i0);
            fr0 = p[0]; fr1 = p[1];
            mreg = mptr[s + 1];
        }

        // column max for this thread's batch column (combine 3 wave partials)
        const float m = fmaxf(Pm[n], fmaxf(Pm[NB + n], Pm[2 * NB + n]));

        // Q[j][n] = exp(score[j][n] - m), stored transposed for contiguous B-operand reads
        #pragma unroll
        for (int k = 0; k < 8; ++k)
            Qt[n * QS + i0 + k] = __expf(Sc[(i0 + k) * SS + n] - m);

        __syncthreads();                                 // (b) Qt + Fb[buf] ready

        // R(16x16) = E[block w](16x48) x Q(48x16), K tiled by 4 (12 fp32 WMMAs)
        v8f acc = {0.f, 0.f, 0.f, 0.f, 0.f, 0.f, 0.f, 0.f};
        const int arow = w * 16 + (l & 15);
        #pragma unroll
        for (int k0 = 0; k0 < TAGS; k0 += 4) {
            v2f a = *(const v2f*)&E[arow * ES + k0 + 2 * hi];
            v2f b = *(const v2f*)&Qt[(l & 15) * QS + k0 + 2 * hi];
            acc = __builtin_amdgcn_wmma_f32_16x16x4_f32(
                false, a, false, b, (short)0, acc, false, false);
        }

        // epilogue: new = f + m + log(r); mask blend; write score; fold column max
        float pm = NEGV;
        const bool msk = (mcur != 0.0f);
        #pragma unroll
        for (int v = 0; v < 8; ++v) {
            const int row = w * 16 + v + 8 * hi;         // D layout: M = v + 8*(lane>=16)
            const float oldv = Sc[row * SS + n];
            const float nv = Fb[buf][row * SS + n] + m + fast_log(clamp_lo(acc[v]));
            const float val = msk ? nv : oldv;
            Sc[row * SS + n] = val;
            pm = fmaxf(pm, val);
        }
        pm = fmaxf(pm, __shfl_xor(pm, 16, 32));          // combine half-waves (same column n)
        if (l < 16) Pm[w * NB + n] = pm;
    }

    __syncthreads();
    // final: out[b] = LSE_j(score[j][b] + trans[STOP, j])
    if (t < NB) {
        float mx = -3.0e38f;
        for (int j = 0; j < TAGS; ++j)
            mx = fmaxf(mx, Sc[j * SS + t] + trans[STOP_IDX * TAGS + j]);
        float sum = 0.0f;
        for (int j = 0; j < TAGS; ++j)
            sum += __expf(Sc[j * SS + t] + trans[STOP_IDX * TAGS + j] - mx);
        out[nbase + t] = mx + fast_log(clamp_lo(sum));
    }
}

extern "C" void kernel_launch(void* const* d_in, const int* in_sizes, int n_in,
                              void* d_out, int out_size, void* d_ws, size_t ws_size,
                              hipStream_t stream) {
    (void)n_in; (void)d_ws; (void)ws_size; (void)out_size;
    const float* feats = (const float*)d_in[0];   // [B, SEQ, TAGS] fp32
    const float* trans = (const float*)d_in[1];   // [TAGS, TAGS]   fp32
    const float* masks = (const float*)d_in[2];   // [B, SEQ]       fp32
    float* out = (float*)d_out;                   // [B]            fp32

    const int B = in_sizes[0] / (SEQ * TAGS);     // 1024
    const int blocks = B / NB;                    // 64 groups, 3 waves each
    crf_fwd_wmma<<<blocks, 96, 0, stream>>>(feats, trans, masks, out);
}